// Input_Net_64072322122340
// MI455X (gfx1250) — compile-verified
//
#include <hip/hip_runtime.h>

#define NT    8192
#define NSRC  8192
#define BATCH 8
#define EDIM  32
#define NH    10

typedef __attribute__((ext_vector_type(2))) float v2f;
typedef __attribute__((ext_vector_type(8))) float v8f;

__device__ __forceinline__ bool dless(float ad, int ai, float bd, int bi) {
    return (ad < bd) || (ad == bd && ai < bi);
}

// ---------------------------------------------------------------------------
// Kernel 1: per-row top-NH over d = sqrt(lon^2 + lat^2). One block per t-row.
// Each thread keeps a sorted top-10 in registers, then LDS tree-merge.
// ---------------------------------------------------------------------------
__global__ void __launch_bounds__(256) topk_kernel(
    const float* __restrict__ rel_lon, const float* __restrict__ rel_lat,
    float* __restrict__ out_dist, float* __restrict__ out_lon,
    float* __restrict__ out_lat, int* __restrict__ ws_idx)
{
    __shared__ float sd[256 * NH];
    __shared__ int   si[256 * NH];
    const int tid = threadIdx.x;
    const int t   = blockIdx.x;
    const float* lon = rel_lon + (size_t)t * NSRC;
    const float* lat = rel_lat + (size_t)t * NSRC;

    float d[NH]; int ix[NH];
#pragma unroll
    for (int k = 0; k < NH; ++k) { d[k] = 3.4e38f; ix[k] = 0x7fffffff; }

    for (int j = 0; j < NSRC / 256; ++j) {
        const int s = tid + j * 256;
        const float lo = lon[s], la = lat[s];
        const float val = sqrtf(lo * lo + la * la);
        if (dless(val, s, d[NH - 1], ix[NH - 1])) {
            float cv = val; int ci = s;
#pragma unroll
            for (int p = 0; p < NH; ++p) {          // bubble-insert, reg-only
                const bool sw = dless(cv, ci, d[p], ix[p]);
                const float td = d[p]; const int ti = ix[p];
                if (sw) { d[p] = cv; ix[p] = ci; cv = td; ci = ti; }
            }
        }
    }
#pragma unroll
    for (int k = 0; k < NH; ++k) { sd[tid * NH + k] = d[k]; si[tid * NH + k] = ix[k]; }

    for (int stride = 128; stride >= 1; stride >>= 1) {
        __syncthreads();
        if (tid < stride) {
            // two-pointer merge of sorted lists tid and tid+stride (reads LDS)
            float md[NH]; int mi[NH];
            int i = 0, j = 0;
            const int a0 = tid * NH, b0 = (tid + stride) * NH;
#pragma unroll
            for (int k = 0; k < NH; ++k) {
                const float av = sd[a0 + i], bv = sd[b0 + j];
                const int   aj = si[a0 + i], bj = si[b0 + j];
                if (dless(av, aj, bv, bj)) { md[k] = av; mi[k] = aj; ++i; }
                else                       { md[k] = bv; mi[k] = bj; ++j; }
            }
#pragma unroll
            for (int k = 0; k < NH; ++k) { sd[a0 + k] = md[k]; si[a0 + k] = mi[k]; }
        }
    }
    __syncthreads();
    if (tid < NH) {
        const float dv = sd[tid];
        const int   iv = si[tid];
        out_dist[t * NH + tid] = dv;
        out_lon [t * NH + tid] = lon[iv];
        out_lat [t * NH + tid] = lat[iv];
        ws_idx  [t * NH + tid] = iv;
    }
}

// ---------------------------------------------------------------------------
// Kernel 2: inv_rowsum[t] = 1 / sum_s w_raw(t,s); w_raw = 1/(sqrt(d2)+eps)
// coords are 64 KB each -> L2 resident; this is compute-only.
// ---------------------------------------------------------------------------
__global__ void __launch_bounds__(256) rowsum_kernel(
    const float* __restrict__ cs, const float* __restrict__ ct,
    float* __restrict__ inv_rowsum)
{
    __shared__ float red[256];
    const int tid = threadIdx.x;
    const int t   = blockIdx.x;
    const float ctx = ct[2 * t], cty = ct[2 * t + 1];
    float acc = 0.0f;
    for (int j = 0; j < NSRC / 256; ++j) {
        const int s = tid + j * 256;
        const float dx = ctx - cs[2 * s];
        const float dy = cty - cs[2 * s + 1];
        const float d2 = fmaxf(dx * dx + dy * dy, 1e-12f);
        acc += 1.0f / (sqrtf(d2) + 1e-10f);
    }
    red[tid] = acc;
    __syncthreads();
    for (int stride = 128; stride >= 1; stride >>= 1) {
        if (tid < stride) red[tid] += red[tid + stride];
        __syncthreads();
    }
    if (tid == 0) inv_rowsum[t] = 1.0f / red[0];
}

// ---------------------------------------------------------------------------
// Kernel 3: x_nearest[b,t,h,:] = x[b, idx[t,h], :]  (float4 copies)
// ---------------------------------------------------------------------------
__global__ void __launch_bounds__(256) gather_kernel(
    const float* __restrict__ x, const int* __restrict__ ws_idx,
    float* __restrict__ out_xn)
{
    const int gid = blockIdx.x * 256 + threadIdx.x;   // BATCH*NT*NH*8 total
    const int f4 = gid & 7;
    const int q  = gid >> 3;
    const int h  = q % NH;
    const int t  = (q / NH) % NT;
    const int b  = q / (NH * NT);
    const int s  = ws_idx[t * NH + h];
    const float4 v = *(const float4*)(x + (size_t)b * NSRC * EDIM + (size_t)s * EDIM + f4 * 4);
    *(float4*)(out_xn + (((size_t)(b * NT + t) * NH + h) * EDIM) + f4 * 4) = v;
}

// ---------------------------------------------------------------------------
// Kernel 4: x_inter = diag(inv_rowsum) * W_raw @ X   via V_WMMA_F32_16X16X4_F32
//   M = t (8192), K = s (8192), N = b*32+e (256).
//   A (16x4 f32, W tile) is generated on the fly from coords into LDS once per
//   workgroup and shared by all 8 waves; each wave owns two 16x16 C tiles.
//   f32 A layout (ISA 7.12.2): lanes 0-15 -> K=kb+0/1 with kb=0, lanes 16-31 kb=2.
// ---------------------------------------------------------------------------
__global__ void __launch_bounds__(256) xinter_kernel(
    const float* __restrict__ x, const float* __restrict__ cs,
    const float* __restrict__ ct, const float* __restrict__ inv_rowsum,
    float* __restrict__ out_xi)
{
    __shared__ float As[64 * 16];   // [s_local][m], 4 KB
    const int tid  = threadIdx.x;
    const int lane = tid & 31;
    const int wave = tid >> 5;
    const int t0   = blockIdx.x * 16;
    const int m    = lane & 15;
    const int kb   = (lane >> 4) * 2;

    // B-operand per-lane base pointers (two 16-col tiles per wave)
    const int n0 = wave * 32 + m;
    const int n1 = n0 + 16;
    const float* xp0 = x + (size_t)(n0 >> 5) * (NSRC * EDIM) + (n0 & 31);
    const float* xp1 = x + (size_t)(n1 >> 5) * (NSRC * EDIM) + (n1 & 31);

    // A-fill assignment: thread fills column m_fill at s_locals sl_base+16i
    const int m_fill  = tid & 15;
    const int sl_base = tid >> 4;
    const float ctx = ct[2 * (t0 + m_fill)];
    const float cty = ct[2 * (t0 + m_fill) + 1];

    v8f c0 = {}; v8f c1 = {};

    for (int s0 = 0; s0 < NSRC; s0 += 64) {
#pragma unroll
        for (int i = 0; i < 4; ++i) {               // cooperative W-tile gen
            const int sl = sl_base + i * 16;
            const float dx = ctx - cs[2 * (s0 + sl)];
            const float dy = cty - cs[2 * (s0 + sl) + 1];
            const float d2 = fmaxf(dx * dx + dy * dy, 1e-12f);
            As[sl * 16 + m_fill] = 1.0f / (sqrtf(d2) + 1e-10f);
        }
        __syncthreads();
        const float* xs0 = xp0 + (size_t)s0 * EDIM;
        const float* xs1 = xp1 + (size_t)s0 * EDIM;
#pragma unroll
        for (int kk = 0; kk < 16; ++kk) {
            const int kA = kk * 4 + kb;
            v2f a, b0, b1;
            a.x  = As[kA * 16 + m];
            a.y  = As[(kA + 1) * 16 + m];
            b0.x = xs0[kA * EDIM];
            b0.y = xs0[(kA + 1) * EDIM];
            b1.x = xs1[kA * EDIM];
            b1.y = xs1[(kA + 1) * EDIM];
            c0 = __builtin_amdgcn_wmma_f32_16x16x4_f32(false, a, false, b0,
                                                       (short)0, c0, false, false);
            c1 = __builtin_amdgcn_wmma_f32_16x16x4_f32(false, a, false, b1,
                                                       (short)0, c1, false, false);
        }
        __syncthreads();
    }

    // scale by inv_rowsum (normalization factored out of the GEMM) and store
    const int half = lane >> 4;
#pragma unroll
    for (int v = 0; v < 8; ++v) {
        const int row = t0 + v + (half ? 8 : 0);
        const float invr = inv_rowsum[row];
        out_xi[(size_t)(n0 >> 5) * (NT * EDIM) + (size_t)row * EDIM + (n0 & 31)] = c0[v] * invr;
        out_xi[(size_t)(n1 >> 5) * (NT * EDIM) + (size_t)row * EDIM + (n1 & 31)] = c1[v] * invr;
    }
}

// ---------------------------------------------------------------------------
extern "C" void kernel_launch(void* const* d_in, const int* in_sizes, int n_in,
                              void* d_out, int out_size, void* d_ws, size_t ws_size,
                              hipStream_t stream)
{
    const float* x       = (const float*)d_in[0];
    const float* rel_lon = (const float*)d_in[1];
    const float* rel_lat = (const float*)d_in[2];
    const float* cs      = (const float*)d_in[3];
    const float* ct      = (const float*)d_in[4];

    float* out = (float*)d_out;
    // outputs concatenated flat, in reference return order
    float* out_xn  = out;                 // (8,8192,10,32) = 20,971,520
    float* out_xi  = out + 20971520;      // (8,8192,32)    =  2,097,152
    float* out_id  = out + 23068672;      // (8192,10)
    float* out_lon = out + 23150592;      // (8192,10)
    float* out_lat = out + 23232512;      // (8192,10)

    int*   ws_idx     = (int*)d_ws;                                    // NT*NH ints
    float* inv_rowsum = (float*)((char*)d_ws + (size_t)NT * NH * 4);   // NT floats

    topk_kernel  <<<NT,       256, 0, stream>>>(rel_lon, rel_lat, out_id, out_lon, out_lat, ws_idx);
    rowsum_kernel<<<NT,       256, 0, stream>>>(cs, ct, inv_rowsum);
    gather_kernel<<<(BATCH * NT * NH * 8) / 256, 256, 0, stream>>>(x, ws_idx, out_xn);
    xinter_kernel<<<NT / 16,  256, 0, stream>>>(x, cs, ct, inv_rowsum, out_xi);
}